// BBox3dProjector_47648367181954
// MI455X (gfx1250) — compile-verified
//
#include <hip/hip_runtime.h>
#include <hip/hip_bf16.h>
#include <math.h>

typedef __attribute__((ext_vector_type(2))) float v2f;
typedef __attribute__((ext_vector_type(8))) float v8f;

#define BOXES_PER_BLOCK 32
#define NTHREADS 256

#if __has_builtin(__builtin_amdgcn_wmma_f32_16x16x4_f32)
#define HAVE_WMMA_F32_4 1
#else
#define HAVE_WMMA_F32_4 0
#endif

// Low 32 bits of a generic pointer to LDS == group-segment byte offset (ISA 10.2).
__device__ __forceinline__ unsigned lds_offset(const void* p) {
    return (unsigned)(size_t)p;
}

__global__ __launch_bounds__(NTHREADS)
void BBox3dProjector_kernel(const float* __restrict__ bbox,   // N x 7
                            const float* __restrict__ p2,     // 3 x 4
                            float* __restrict__ out_pts,      // N*8*3
                            float* __restrict__ out_homo,     // N*8*3 (int16 values as float)
                            float* __restrict__ out_theta,    // N
                            int N)
{
    __shared__ __align__(16) float sraw[BOXES_PER_BLOCK * 7];   // staged input rows
    __shared__ float sbox[BOXES_PER_BLOCK * 10];                // per-box precompute

    const int t = threadIdx.x;
    const int boxbase = blockIdx.x * BOXES_PER_BLOCK;
    const bool full = (boxbase + BOXES_PER_BLOCK) <= N;

    // ---- Stage 32 rows (224 floats = 56 x b128) global -> LDS via async DMA ----
    if (full) {
        if (t < 56) {
            const float* gsrc = bbox + (size_t)boxbase * 7 + (size_t)t * 4;
            unsigned long long gaddr = (unsigned long long)(size_t)gsrc;
            unsigned ldso = lds_offset(&sraw[t * 4]);
            asm volatile("global_load_async_to_lds_b128 %0, %1, off"
                         :: "v"(ldso), "v"(gaddr) : "memory");
        }
#if __has_builtin(__builtin_amdgcn_s_wait_asynccnt)
        __builtin_amdgcn_s_wait_asynccnt(0);
#else
        asm volatile("s_wait_asynccnt 0x0" ::: "memory");
#endif
    } else {
        int nfloat = (N - boxbase) * 7;
        for (int i = t; i < nfloat; i += NTHREADS)
            sraw[i] = bbox[(size_t)boxbase * 7 + i];
    }
    __syncthreads();

    // ---- Uniform projection matrix (scalar loads) ----
    const float p00 = p2[0], p01 = p2[1],  p02 = p2[2],  p03 = p2[3];
    const float p10 = p2[4], p11 = p2[5],  p12 = p2[6],  p13 = p2[7];
    const float p20 = p2[8], p21 = p2[9],  p22 = p2[10], p23 = p2[11];
    const float offs = p03 / p00;

    // ---- Phase 1: one lane per box does the transcendentals, once ----
    if (t < BOXES_PER_BLOCK && (boxbase + t) < N) {
        const float* r = &sraw[t * 7];
        float cx = r[0], cy = r[1], cz = r[2];
        float w  = r[3], h  = r[4], l  = r[5];
        float alpha = r[6];

        float theta = alpha + atan2f(cx + offs, cz);
        float st, ct;
        sincosf(theta, &st, &ct);
        float hh = 0.5f * h, hl = 0.5f * l;

        float* s = &sbox[t * 10];
        s[0] = cx; s[1] = cy; s[2] = cz; s[3] = w;
        s[4] = hh * st;
        s[5] = hl * ct;
        s[6] = hh * ct;
        s[7] = hl * st;

        out_theta[boxbase + t] = theta;   // coalesced
    }
    __syncthreads();

    // ---- Phase 2: corner coordinates, one corner per lane ----
    const int b = t >> 3;           // box within block
    const int k = t & 7;            // corner index
    const float* s = &sbox[b * 10];
    float cx = s[0], cy = s[1], cz = s[2], w = s[3];
    float a1 = s[4], a2 = s[5], a3 = s[6], a4 = s[7];

    float I  = (k < 4) ? 1.0f : -1.0f;
    float J  = ((k >> 1) & 1) ? -1.0f : 1.0f;
    float Kk = (float)(k & 1);
    float IJ = I * J;

    float px = cx - I * a1 + IJ * a2;
    float pz = cz - I * a3 - IJ * a4;
    float py = cy - Kk * w;

    const float ca = 0.45399049973954675f;   // cos(63 deg)
    const float sa = 0.8910065241883679f;    // sin(63 deg)

#if HAVE_WMMA_F32_4
    if (full) {
        // --- WMMA path: D = A(16x4) x B(4x16); EXEC all-1s on this uniform branch ---
        const int lane = t & 31;
        const int m    = lane & 15;   // row of A / column of B
        const int hi   = lane >> 4;   // K-half selector

        // A_rot: rotation about X padded to 16x4 (rows 0..2)
        v2f arot = {0.0f, 0.0f};
        if (hi == 0) {                       // holds K=0 (x), K=1 (y)
            if      (m == 0) { arot.x = 1.0f; }
            else if (m == 1) { arot.y = ca;   }
            else if (m == 2) { arot.y = sa;   }
        } else {                             // holds K=2 (z), K=3 (unused)
            if      (m == 1) { arot.x = -sa;  }
            else if (m == 2) { arot.x = ca;   }
        }

        // A_p2: full 3x4 projection (K=3 column adds P2[:,3] via homogeneous 1)
        v2f ap2 = {0.0f, 0.0f};
        if      (m == 0) { ap2.x = hi ? p02 : p00; ap2.y = hi ? p03 : p01; }
        else if (m == 1) { ap2.x = hi ? p12 : p10; ap2.y = hi ? p13 : p11; }
        else if (m == 2) { ap2.x = hi ? p22 : p20; ap2.y = hi ? p23 : p21; }

        const bool lo = (hi == 0);
        const size_t wave_pt_base = (size_t)boxbase * 8 + (size_t)(t >> 5) * 32;
        const v8f zero8 = {0.0f, 0.0f, 0.0f, 0.0f, 0.0f, 0.0f, 0.0f, 0.0f};

#pragma unroll
        for (int h = 0; h < 2; ++h) {
            // B_rot for points [16h, 16h+16): VGPR0 = {px | pz}, VGPR1 = {py | 0}
            int src = m + 16 * h;
            float bx = __shfl(px, src, 32);
            float by = __shfl(py, src, 32);
            float bz = __shfl(pz, src, 32);
            v2f brot;
            brot.x = lo ? bx : bz;
            brot.y = lo ? by : 0.0f;

            v8f drot = __builtin_amdgcn_wmma_f32_16x16x4_f32(
                false, arot, false, brot, (short)0, zero8, false, false);

            // lanes 0-15: rows 0..2 = rotated point of column m
            float rx = drot[0];
            float ry = drot[1];
            float rz = fmaxf(drot[2], 1.0f);

            // B_p2: VGPR0 = {rx | rz}, VGPR1 = {ry | 1}
            float rzn = __shfl(rz, m, 32);   // lanes>=16 fetch rz from lane-16
            v2f bp2;
            bp2.x = lo ? rx : rzn;
            bp2.y = lo ? ry : 1.0f;

            v8f dp2 = __builtin_amdgcn_wmma_f32_16x16x4_f32(
                false, ap2, false, bp2, (short)0, zero8, false, false);

            if (lo) {
                float hx = dp2[0], hy = dp2[1], hz = dp2[2];
                float q0 = hx / hz;
                float q1 = hy / hz;
                float q2 = hz / hz;          // exactly 1.0, like the reference

                size_t g = wave_pt_base + (size_t)(16 * h) + (size_t)m;
                size_t o = g * 3;
                out_pts[o + 0] = rx;
                out_pts[o + 1] = ry;
                out_pts[o + 2] = rz;
                out_homo[o + 0] = (float)(short)(int)q0;
                out_homo[o + 1] = (float)(short)(int)q1;
                out_homo[o + 2] = (float)(short)(int)q2;
            }
        }
        return;
    }
#endif

    // --- Scalar path (tail blocks, or WMMA builtin unavailable) ---
    const int gbox = boxbase + b;
    if (gbox < N) {
        float rx = px;
        float ry = ca * py - sa * pz;
        float rz = sa * py + ca * pz;
        rz = fmaxf(rz, 1.0f);

        float hx = p00 * rx + p01 * ry + p02 * rz + p03;
        float hy = p10 * rx + p11 * ry + p12 * rz + p13;
        float hz = p20 * rx + p21 * ry + p22 * rz + p23;
        float q0 = hx / hz;
        float q1 = hy / hz;
        float q2 = hz / hz;

        size_t o = ((size_t)gbox * 8 + (size_t)k) * 3;
        out_pts[o + 0] = rx;
        out_pts[o + 1] = ry;
        out_pts[o + 2] = rz;
        out_homo[o + 0] = (float)(short)(int)q0;
        out_homo[o + 1] = (float)(short)(int)q1;
        out_homo[o + 2] = (float)(short)(int)q2;
    }
}

extern "C" void kernel_launch(void* const* d_in, const int* in_sizes, int n_in,
                              void* d_out, int out_size, void* d_ws, size_t ws_size,
                              hipStream_t stream) {
    const float* bbox = (const float*)d_in[0];
    const float* p2   = (const float*)d_in[1];
    int N = in_sizes[0] / 7;

    float* out       = (float*)d_out;
    float* out_pts   = out;                          // N*24
    float* out_homo  = out + (size_t)N * 24;         // N*24
    float* out_theta = out + (size_t)N * 48;         // N

    int blocks = (N + BOXES_PER_BLOCK - 1) / BOXES_PER_BLOCK;
    BBox3dProjector_kernel<<<blocks, NTHREADS, 0, stream>>>(
        bbox, p2, out_pts, out_homo, out_theta, N);
}